// GaussianBlur3D_46308337386351
// MI455X (gfx1250) — compile-verified
//
#include <hip/hip_runtime.h>

// CDNA5 / gfx1250: separable 3D Gaussian blur. Each 1D pass = banded 16x16
// GEMM tiles on V_WMMA_F32_16X16X4_F32 (full f32, matches reference math).
// Each wave sweeps a full 192-long line with a rolling A-fragment buffer
// (12-position tile overlap reused). Reflection handling is peeled out of
// the steady-state loop: only tile 0's preload and tile 11's refill touch
// the edges, so the hot loop is branch-free with immediate-offset loads.

typedef float v2f __attribute__((ext_vector_type(2)));
typedef float v8f __attribute__((ext_vector_type(8)));

#define NDIM 192
#define KS   11
#define RAD  5
#define TILES_AXIS (NDIM / 16)                       // 12 tiles per line
#define NLINES (2 * NDIM * NDIM)                     // 73728 lines per pass
#define LINE_TILES (NLINES / 16)                     // 4608 wave-jobs per pass
#define WAVES_PER_BLOCK 8

__device__ __forceinline__ int refl(int p) {
  p = (p < 0) ? -p : p;                               // jnp reflect: -1 -> 1
  return (p > (NDIM - 1)) ? (2 * (NDIM - 1) - p) : p; // N -> N-2
}

// INNER = element stride along the convolution axis. Lines (the other two
// axes flattened) are contiguous in memory whenever INNER > 1.
template <int INNER>
__global__ __launch_bounds__(256) void blur_pass(
    const float* __restrict__ src, float* __restrict__ dst,
    const float* __restrict__ sigma_p) {
  __shared__ float s_w[16];
  if (threadIdx.x == 0) {
    float s = fabsf(sigma_p[0]) + 1e-6f;
    float inv = -1.0f / (2.0f * s * s);
    float tmp[KS];
    float sum = 0.0f;
#pragma unroll
    for (int t = 0; t < KS; ++t) {
      float r = (float)t - (float)RAD;
      tmp[t] = expf(r * r * inv);
      sum += tmp[t];
    }
#pragma unroll
    for (int t = 0; t < KS; ++t) s_w[t] = tmp[t] / sum;
#pragma unroll
    for (int t = KS; t < 16; ++t) s_w[t] = 0.0f;      // zero pad for OOB band
  }
  __syncthreads();

  const int lane = threadIdx.x & 31;
  const int wave = threadIdx.x >> 5;
  const int n    = lane & 15;        // B/D column (output position in tile)
  const int half = lane >> 4;        // lane-half selects K sub-rows / D row+8
  const int koff = half * 2;

  // B fragments: banded weight matrix B[j][n] = w[j-n], j = K index.
  v2f bfrag[7];
#pragma unroll
  for (int k4 = 0; k4 < 7; ++k4) {
#pragma unroll
    for (int e = 0; e < 2; ++e) {
      int idx = 4 * k4 + koff + e - n;                // j - n
      int ci = ((unsigned)idx <= 10u) ? idx : 15;     // s_w[15] == 0
      bfrag[k4][e] = s_w[ci];
    }
  }

  const int lineTile = blockIdx.x * WAVES_PER_BLOCK + wave;
  if (lineTile >= LINE_TILES) return;  // wave-uniform; grid sized exactly

  const int line0 = lineTile * 16;

  // Row stride between consecutive lines of a tile.
  constexpr long RSTR = (INNER == 1) ? (long)NDIM : 1L;
  // Base of line0 (line0 is a multiple of 16; INNER is a multiple of 16 when
  // >1, so lines line0..line0+15 never cross an outer-dim boundary).
  const long lb0 =
      (long)(line0 / INNER) * (long)(NDIM * INNER) + (long)(line0 % INNER);

  const float* sA = src + lb0 + (long)n * RSTR;       // lane's A-row base
  float* dO = dst + lb0 + (long)(half * 8) * RSTR + (long)n * INNER;

  v2f afr[7];

  auto mma7 = [&]() -> v8f {
    v8f acc = {};
#pragma unroll
    for (int j = 0; j < 7; ++j) {
      acc = __builtin_amdgcn_wmma_f32_16x16x4_f32(
          false, afr[j], false, bfrag[j], (short)0, acc, false, false);
    }
    return acc;
  };

  // ---- Preload tile 0 (low-edge reflection) ----
#pragma unroll
  for (int j = 0; j < 7; ++j) {
#pragma unroll
    for (int e = 0; e < 2; ++e) {
      int p = -RAD + 4 * j + koff + e;                // p in [-5, 22]
      afr[j][e] = sA[(long)refl(p) * INNER];
    }
  }

  // Steady-state refill base: koff folded in -> immediate offsets below.
  const float* cur = sA + (long)koff * INNER;

  // ---- Tiles 0..9: branch-free body + in-bounds refill for tile t+1 ----
#pragma unroll 1
  for (int t = 0; t < TILES_AXIS - 2; ++t) {
    v8f acc = mma7();
#pragma unroll
    for (int r = 0; r < 8; ++r) dO[(long)r * RSTR] = acc[r];
    dO += (long)16 * INNER;

    // Refill chunks 3..6 of tile t+1: absolute p = 16t + koff + 23+4jj+e,
    // always in [23, 182] -> no reflection.
#pragma unroll
    for (int j = 0; j < 3; ++j) afr[j] = afr[j + 4];
#pragma unroll
    for (int jj = 0; jj < 4; ++jj) {
#pragma unroll
      for (int e = 0; e < 2; ++e)
        afr[3 + jj][e] = cur[(long)(23 + 4 * jj + e) * INNER];
    }
    cur += (long)16 * INNER;
  }

  // ---- Tile 10: body, then high-edge-reflected refill for tile 11 ----
  {
    v8f acc = mma7();
#pragma unroll
    for (int r = 0; r < 8; ++r) dO[(long)r * RSTR] = acc[r];
    dO += (long)16 * INNER;

#pragma unroll
    for (int j = 0; j < 3; ++j) afr[j] = afr[j + 4];
    const int pos0n = (TILES_AXIS - 1) * 16;          // 176
#pragma unroll
    for (int j = 3; j < 7; ++j) {
#pragma unroll
      for (int e = 0; e < 2; ++e) {
        int p = pos0n - RAD + 4 * j + koff + e;       // p in [183, 198]
        afr[j][e] = sA[(long)refl(p) * INNER];
      }
    }
  }

  // ---- Tile 11 ----
  {
    v8f acc = mma7();
#pragma unroll
    for (int r = 0; r < 8; ++r) dO[(long)r * RSTR] = acc[r];
  }
}

extern "C" void kernel_launch(void* const* d_in, const int* in_sizes, int n_in,
                              void* d_out, int out_size, void* d_ws,
                              size_t ws_size, hipStream_t stream) {
  const float* x     = (const float*)d_in[0];
  const float* sigma = (const float*)d_in[1];
  float* out = (float*)d_out;
  float* ws  = (float*)d_ws;   // needs 2*192^3 floats = 56.6 MB

  dim3 block(256);
  dim3 grid(LINE_TILES / WAVES_PER_BLOCK);            // 576, exact

  // Pass 1: conv along W (stride 1):      x  -> out
  blur_pass<1><<<grid, block, 0, stream>>>(x, out, sigma);
  // Pass 2: conv along H (stride 192):    out -> ws
  blur_pass<NDIM><<<grid, block, 0, stream>>>(out, ws, sigma);
  // Pass 3: conv along D (stride 192^2):  ws -> out
  blur_pass<NDIM * NDIM><<<grid, block, 0, stream>>>(ws, out, sigma);
}